// ProbAttention_47622597378699
// MI455X (gfx1250) — compile-verified
//
#include <hip/hip_runtime.h>
#include <math.h>

#define Bn 4
#define Ln 2048
#define Hn 8
#define Dn 64
#define Un 40
#define NC 16            // cumsum chunks
#define CHK 128          // positions per chunk
#define SCALE 0.125f     // 1/sqrt(64)
#define NEG_BIG -1.0e30f

typedef __attribute__((ext_vector_type(2))) float v2f;
typedef __attribute__((ext_vector_type(8))) float v8f;

// ---------------------------------------------------------------------------
// Phase 1: M[b,h,q] = max_s(Q.K_sample) - sum_s(Q.K_sample)/L
// One wave32 per (b,h,q); lane holds dims 2*lane, 2*lane+1.
// ---------------------------------------------------------------------------
__global__ __launch_bounds__(256) void pa_sample_m(
    const float* __restrict__ Q, const float* __restrict__ K,
    const int* __restrict__ samp, float* __restrict__ Mout) {
  int wid  = blockIdx.x * 8 + (threadIdx.x >> 5);
  int lane = threadIdx.x & 31;
  int q = wid % Ln;
  int h = (wid / Ln) % Hn;
  int b = wid / (Ln * Hn);

  const float* qrow  = Q + (((size_t)b * Ln + q) * Hn + h) * Dn;
  float qx = qrow[2 * lane], qy = qrow[2 * lane + 1];
  const float* kbase = K + ((size_t)b * Ln * Hn + h) * Dn;

  float mx = NEG_BIG, sm = 0.f;
  for (int s = 0; s < Un; ++s) {
    int k = samp[q * Un + s];
    const float* krow = kbase + (size_t)k * Hn * Dn;
    float d = qx * krow[2 * lane] + qy * krow[2 * lane + 1];
    for (int off = 16; off; off >>= 1) d += __shfl_xor(d, off, 32);
    mx = fmaxf(mx, d);
    sm += d;
  }
  if (lane == 0) Mout[((size_t)b * Hn + h) * Ln + q] = mx - sm / (float)Ln;
}

// ---------------------------------------------------------------------------
// Phase 2: per (b,h) top-40 of M (iterative block argmax, ties -> lowest idx)
// ---------------------------------------------------------------------------
__global__ __launch_bounds__(256) void pa_topk(
    const float* __restrict__ Mv, int* __restrict__ top) {
  __shared__ float vals[Ln];
  __shared__ float rv[256];
  __shared__ int   ri[256];
  int tid = threadIdx.x, bh = blockIdx.x;

  for (int i = tid; i < Ln; i += 256) vals[i] = Mv[(size_t)bh * Ln + i];
  __syncthreads();

  for (int it = 0; it < Un; ++it) {
    float best = -3.4e38f; int bi = 0;
    for (int i = tid; i < Ln; i += 256) {
      float v = vals[i];
      if (v > best) { best = v; bi = i; }
    }
    rv[tid] = best; ri[tid] = bi;
    __syncthreads();
    for (int off = 128; off; off >>= 1) {
      if (tid < off) {
        float ov = rv[tid + off]; int oi = ri[tid + off];
        if (ov > rv[tid] || (ov == rv[tid] && oi < ri[tid])) { rv[tid] = ov; ri[tid] = oi; }
      }
      __syncthreads();
    }
    if (tid == 0) { top[bh * Un + it] = ri[0]; vals[ri[0]] = -3.4e38f; }
    __syncthreads();
  }
}

// ---------------------------------------------------------------------------
// Phase 3: cumsum(V, seq) into d_out (chunk sums -> exclusive scan -> write)
// ---------------------------------------------------------------------------
__global__ __launch_bounds__(64) void pa_csum_chunks(
    const float* __restrict__ V, float* __restrict__ cs) {
  int blk = blockIdx.x, c = blk % NC, bh = blk / NC;
  int b = bh / Hn, h = bh % Hn, d = threadIdx.x;
  const float* Vb = V + ((size_t)b * Ln * Hn + h) * Dn + d;
  float s = 0.f;
  int base = c * CHK;
  for (int i = 0; i < CHK; ++i) s += Vb[(size_t)(base + i) * Hn * Dn];
  cs[((size_t)bh * NC + c) * Dn + d] = s;
}

__global__ __launch_bounds__(256) void pa_csum_scan(float* cs) {
  int t = blockIdx.x * blockDim.x + threadIdx.x;   // over B*H*D
  int bh = t / Dn, d = t % Dn;
  float run = 0.f;
  for (int c = 0; c < NC; ++c) {
    size_t idx = ((size_t)bh * NC + c) * Dn + d;
    float v = cs[idx];
    cs[idx] = run;                                  // exclusive prefix
    run += v;
  }
}

__global__ __launch_bounds__(64) void pa_csum_write(
    const float* __restrict__ V, const float* __restrict__ cs,
    float* __restrict__ out) {
  int blk = blockIdx.x, c = blk % NC, bh = blk / NC;
  int b = bh / Hn, h = bh % Hn, d = threadIdx.x;
  const float* Vb = V + ((size_t)b * Ln * Hn + h) * Dn + d;
  float*       ob = out + ((size_t)b * Ln * Hn + h) * Dn + d;
  float run = cs[((size_t)bh * NC + c) * Dn + d];
  int base = c * CHK;
  for (int i = 0; i < CHK; ++i) {
    run += Vb[(size_t)(base + i) * Hn * Dn];
    ob[(size_t)(base + i) * Hn * Dn] = run;
  }
}

// ---------------------------------------------------------------------------
// Phase 4: WMMA attention for the 40 selected queries per (b,h).
// Block = (b,h), 256 threads = 8 waves = 4 row-tiles x 2 column groups.
// Scores: S(64x2048) = Qtop(64x64) x K^T via v_wmma_f32_16x16x4_f32 (16 steps)
// P.V   : A(64x64)  += P(16x16) x V(16x64) via 16 more WMMAs per key tile.
// ---------------------------------------------------------------------------
__global__ __launch_bounds__(256) void pa_attn(
    const float* __restrict__ Q, const float* __restrict__ K,
    const float* __restrict__ V, const int* __restrict__ top,
    float* __restrict__ out) {
  __shared__ float Qs[64][68];
  __shared__ float Ptile[8][16][17];
  __shared__ float Abuf[64][68];
  __shared__ int   cut[64];
  __shared__ float pmax[2][64], psum[2][64];
  __shared__ float rowMax[64], rowSum[64];

  int tid = threadIdx.x;
  int bh = blockIdx.x, b = bh / Hn, h = bh % Hn;
  int wv = tid >> 5, lane = tid & 31, kh = lane >> 4, ln = lane & 15;
  int rt = wv >> 1, cg = wv & 1;          // row tile / column group

  const float* Kb = K + ((size_t)b * Ln * Hn + h) * Dn;
  const float* Vb = V + ((size_t)b * Ln * Hn + h) * Dn;

  if (tid < 64) cut[tid] = (tid < Un) ? top[bh * Un + tid] : -1;
  for (int i = tid; i < 64 * 68; i += 256) (&Abuf[0][0])[i] = 0.f;
  __syncthreads();

  // Gather Q_top rows (pad 40..63 with zeros)
  for (int i = tid; i < 64 * Dn; i += 256) {
    int u = i >> 6, d = i & 63;
    float v = 0.f;
    if (u < Un) v = Q[(((size_t)b * Ln + cut[u]) * Hn + h) * Dn + d];
    Qs[u][d] = v;
  }
  __syncthreads();

  // ----- Pass 1: masked row max -----
  float vmax[8];
  #pragma unroll
  for (int r = 0; r < 8; ++r) vmax[r] = NEG_BIG;

  for (int ct = cg; ct < Ln / 16; ct += 2) {
    const float* Krow = Kb + (size_t)(ct * 16 + ln) * Hn * Dn;
    v8f acc = {};
    #pragma unroll
    for (int kq = 0; kq < 16; ++kq) {
      v2f a, bb;
      a.x  = Qs[rt * 16 + ln][4 * kq + 2 * kh];
      a.y  = Qs[rt * 16 + ln][4 * kq + 2 * kh + 1];
      bb.x = Krow[4 * kq + 2 * kh];
      bb.y = Krow[4 * kq + 2 * kh + 1];
      acc = __builtin_amdgcn_wmma_f32_16x16x4_f32(false, a, false, bb,
                                                  (short)0, acc, false, false);
    }
    int col = ct * 16 + ln;
    #pragma unroll
    for (int r = 0; r < 8; ++r) {
      int row = rt * 16 + r + 8 * kh;
      float s = (col <= cut[row]) ? acc[r] * SCALE : NEG_BIG;
      vmax[r] = fmaxf(vmax[r], s);
    }
  }
  #pragma unroll
  for (int r = 0; r < 8; ++r)
    for (int off = 1; off < 16; off <<= 1)
      vmax[r] = fmaxf(vmax[r], __shfl_xor(vmax[r], off, 32));
  if (ln == 0)
    for (int r = 0; r < 8; ++r) pmax[cg][rt * 16 + r + 8 * kh] = vmax[r];
  __syncthreads();
  if (tid < 64) rowMax[tid] = fmaxf(pmax[0][tid], pmax[1][tid]);
  __syncthreads();

  // ----- Pass 2: exp, row sums, A += P.V -----
  float vsum[8];
  v8f accA[4];
  v8f zz = {};
  #pragma unroll
  for (int r = 0; r < 8; ++r) vsum[r] = 0.f;
  #pragma unroll
  for (int dt = 0; dt < 4; ++dt) accA[dt] = zz;

  for (int ct = cg; ct < Ln / 16; ct += 2) {
    const float* Krow = Kb + (size_t)(ct * 16 + ln) * Hn * Dn;
    v8f acc = {};
    #pragma unroll
    for (int kq = 0; kq < 16; ++kq) {
      v2f a, bb;
      a.x  = Qs[rt * 16 + ln][4 * kq + 2 * kh];
      a.y  = Qs[rt * 16 + ln][4 * kq + 2 * kh + 1];
      bb.x = Krow[4 * kq + 2 * kh];
      bb.y = Krow[4 * kq + 2 * kh + 1];
      acc = __builtin_amdgcn_wmma_f32_16x16x4_f32(false, a, false, bb,
                                                  (short)0, acc, false, false);
    }
    int col = ct * 16 + ln;
    #pragma unroll
    for (int r = 0; r < 8; ++r) {
      int row = rt * 16 + r + 8 * kh;
      float p = 0.f;
      if (col <= cut[row]) p = __expf(acc[r] * SCALE - rowMax[row]);
      vsum[r] += p;
      Ptile[wv][r + 8 * kh][ln] = p;   // C-layout -> LDS
    }
    __syncthreads();
    // Re-read P in A-matrix layout, multiply by V tile (16 keys x 64 dims)
    #pragma unroll
    for (int dt = 0; dt < 4; ++dt) {
      #pragma unroll
      for (int kq = 0; kq < 4; ++kq) {
        v2f a, bb;
        a.x = Ptile[wv][ln][4 * kq + 2 * kh];
        a.y = Ptile[wv][ln][4 * kq + 2 * kh + 1];
        int k0 = ct * 16 + 4 * kq + 2 * kh;
        bb.x = Vb[(size_t)k0 * Hn * Dn + dt * 16 + ln];
        bb.y = Vb[(size_t)(k0 + 1) * Hn * Dn + dt * 16 + ln];
        accA[dt] = __builtin_amdgcn_wmma_f32_16x16x4_f32(false, a, false, bb,
                                                         (short)0, accA[dt], false, false);
      }
    }
    __syncthreads();
  }

  #pragma unroll
  for (int r = 0; r < 8; ++r)
    for (int off = 1; off < 16; off <<= 1)
      vsum[r] += __shfl_xor(vsum[r], off, 32);
  if (ln == 0)
    for (int r = 0; r < 8; ++r) psum[cg][rt * 16 + r + 8 * kh] = vsum[r];

  // Combine the two column groups' A via LDS float atomics (ds_add_f32)
  #pragma unroll
  for (int dt = 0; dt < 4; ++dt)
    #pragma unroll
    for (int r = 0; r < 8; ++r)
      atomicAdd(&Abuf[rt * 16 + r + 8 * kh][dt * 16 + ln], accA[dt][r]);
  __syncthreads();
  if (tid < 64) rowSum[tid] = psum[0][tid] + psum[1][tid];
  __syncthreads();

  // Normalize and scatter over the cumsum output
  for (int i = tid; i < Un * Dn; i += 256) {
    int u = i >> 6, d = i & 63;
    out[(((size_t)b * Ln + cut[u]) * Hn + h) * Dn + d] = Abuf[u][d] / rowSum[u];
  }
}

// ---------------------------------------------------------------------------
extern "C" void kernel_launch(void* const* d_in, const int* in_sizes, int n_in,
                              void* d_out, int out_size, void* d_ws, size_t ws_size,
                              hipStream_t stream) {
  const float* Q   = (const float*)d_in[0];
  const float* K   = (const float*)d_in[1];
  const float* V   = (const float*)d_in[2];
  const int*  samp = (const int*)d_in[3];
  float* out = (float*)d_out;

  char* ws    = (char*)d_ws;
  float* wsM  = (float*)ws;                       // B*H*L floats   (256 KB)
  int*  wsTop = (int*)(ws + 262144);              // B*H*40 ints    (5 KB, padded)
  float* wsCS = (float*)(ws + 262144 + 8192);     // B*H*NC*D floats (128 KB)

  pa_sample_m<<<(Bn * Hn * Ln) / 8, 256, 0, stream>>>(Q, K, samp, wsM);
  pa_topk<<<Bn * Hn, 256, 0, stream>>>(wsM, wsTop);
  pa_csum_chunks<<<Bn * Hn * NC, 64, 0, stream>>>(V, wsCS);
  pa_csum_scan<<<(Bn * Hn * Dn) / 256, 256, 0, stream>>>(wsCS);
  pa_csum_write<<<Bn * Hn * NC, 64, 0, stream>>>(V, wsCS, out);
  pa_attn<<<Bn * Hn, 256, 0, stream>>>(Q, K, V, wsTop, out);
}